// Model_29824252903853
// MI455X (gfx1250) — compile-verified
//
#include <hip/hip_runtime.h>
#include <stdint.h>

// Problem constants (from the reference).
#define N_EVENTS   16
#define N_SAMPLES  32768
#define STEP       256

// Tiling: 128 threads * float4 = 512-float chunk; 4 chunks per block strip.
#define THREADS    128
#define CHUNK      512
#define NCHUNK     4
#define STRIP      (CHUNK * NCHUNK)   // 2048 samples per block

// Native clang vector type (required by __builtin_nontemporal_store).
typedef float v4f __attribute__((ext_vector_type(4)));

// out[b, t] = sum_e [t >= te] * x[b, e, t - te],  te = 256 * idx[b, e].
// Inputs are streamed global->LDS with async-copy (ASYNCcnt) and double
// buffered; each thread consumes exactly the LDS bytes it DMA'd, so no
// workgroup barrier is required -- only s_wait_asynccnt.
__global__ __launch_bounds__(THREADS) void render_gather_kernel(
    const float* __restrict__ x,   // (B, 16, 32768)
    const int*   __restrict__ idx, // (B, 16)
    float*       __restrict__ out) // (B, 32768)
{
    __shared__ float stage[2][N_EVENTS][CHUNK];   // 64 KB

    const int b      = blockIdx.y;
    const int strip0 = blockIdx.x * STRIP;
    const int p4     = threadIdx.x * 4;           // float4 slot within a chunk

    const float* xb = x + (size_t)b * N_EVENTS * N_SAMPLES;

    // Event start times: uniform address -> scalar loads; force wave-uniform.
    int te[N_EVENTS];
#pragma unroll
    for (int e = 0; e < N_EVENTS; ++e)
        te[e] = __builtin_amdgcn_readfirstlane(idx[b * N_EVENTS + e] * STEP);

    // Issue 16 async global->LDS B128 copies for chunk c into buffer (c&1).
    // Source offset is clamped to 0 for not-yet-active lanes (those lanes'
    // data is masked off at accumulate time); this keeps the per-chunk
    // ASYNCcnt increment fixed at 16 so immediate wait thresholds are valid.
    auto issue = [&](int c) {
        const int buf = c & 1;
        const int t   = strip0 + c * CHUNK + p4;
#pragma unroll
        for (int e = 0; e < N_EVENTS; ++e) {
            int off = t - te[e];
            off = off < 0 ? 0 : off;                       // clamp (masked later)
            const float* g = xb + (size_t)e * N_SAMPLES + off;
            unsigned lds = (unsigned)(uintptr_t)&stage[buf][e][p4];
            asm volatile("global_load_async_to_lds_b128 %0, %1, off"
                         :: "v"(lds), "v"((uint64_t)(uintptr_t)g)
                         : "memory");
        }
    };

    // Reduce chunk c from LDS and store (streamed, non-temporal).
    auto accum = [&](int c) {
        const int buf = c & 1;
        const int t0  = strip0 + c * CHUNK;
        const int t   = t0 + p4;
        v4f a = (v4f)(0.f);
#pragma unroll
        for (int e = 0; e < N_EVENTS; ++e) {
            if (te[e] < t0 + CHUNK) {                      // uniform: event touches chunk
                const v4f v = *(const v4f*)&stage[buf][e][p4];
                if (t >= te[e])                            // uniform per float4
                    a += v;
            }
        }
        __builtin_nontemporal_store(
            a, (v4f*)(out + (size_t)b * N_SAMPLES + t));
    };

    // Software pipeline: two chunks in flight (<= 32 outstanding async ops).
    issue(0);
    issue(1);
#pragma unroll
    for (int c = 0; c < NCHUNK; ++c) {
        if (c + 1 < NCHUNK) {
            // Drain the older chunk's 16 ops; keep the newer 16 in flight.
            asm volatile("s_wait_asynccnt 16" ::: "memory");
        } else {
            asm volatile("s_wait_asynccnt 0" ::: "memory");
        }
        accum(c);
        if (c + 2 < NCHUNK) issue(c + 2);
    }
}

extern "C" void kernel_launch(void* const* d_in, const int* in_sizes, int n_in,
                              void* d_out, int out_size, void* d_ws, size_t ws_size,
                              hipStream_t stream) {
    const float* x   = (const float*)d_in[0];
    const int*   idx = (const int*)d_in[1];
    float*       out = (float*)d_out;

    const int B = out_size / N_SAMPLES;            // 64
    dim3 grid(N_SAMPLES / STRIP, B);               // (16, 64)
    render_gather_kernel<<<grid, dim3(THREADS), 0, stream>>>(x, idx, out);
}